// InnerLinearAttention_88450556493923
// MI455X (gfx1250) — compile-verified
//
#include <hip/hip_runtime.h>

typedef __attribute__((ext_vector_type(2))) float v2f;
typedef __attribute__((ext_vector_type(8))) float v8f;

#define NB     4
#define LSEQ   4096
#define NH     16
#define DIM    64
#define PAIRS  (NB * NH)          // 64
#define SPLIT  8                  // s-dim splits per pair in kernel 1
#define SPAN   (LSEQ / SPLIT)     // 512 sequence positions per block
#define SCHUNK 64                 // LDS-staged chunk of the sequence
#define SSTRIDE (NH * DIM)        // 1024 floats between consecutive s

// elu(x) + 1  ==  x>0 ? x+1 : exp(x)
__device__ __forceinline__ float elu1(float x) {
    return x > 0.0f ? x + 1.0f : __expf(x);
}

__global__ __launch_bounds__(256) void zero_ws_kernel(float* __restrict__ ws, int n) {
    int i = blockIdx.x * blockDim.x + threadIdx.x;
    if (i < n) ws[i] = 0.0f;
}

// ---------------------------------------------------------------------------
// Kernel 1: KV[m,d] = sum_s phi(K)[s,d] * V[s,m]   and   Ksum[d] = sum_s phi(K)[s,d]
// grid = PAIRS*SPLIT blocks, 512 threads (16 waves). Each wave owns one 16x16
// tile of the 64x64 KV matrix; reduction over its 512-long s-range in LDS
// chunks of 64, K=4 per v_wmma_f32_16x16x4_f32.
// ---------------------------------------------------------------------------
__global__ __launch_bounds__(512) void kv_reduce_kernel(
    const float* __restrict__ Kg, const float* __restrict__ Vg,
    float* __restrict__ kv_ws, float* __restrict__ ksum_ws)
{
    __shared__ float Kf[SCHUNK][DIM];   // feature-mapped K chunk
    __shared__ float Vr[SCHUNK][DIM];   // raw V chunk

    const int pair  = blockIdx.x >> 3;          // n*NH + h
    const int split = blockIdx.x & 7;
    const int n     = pair >> 4;
    const int h     = pair & 15;
    const int tid   = threadIdx.x;
    const int wave  = tid >> 5;
    const int lane  = tid & 31;
    const int g     = lane >> 4;                // lane-half (K split)
    const int ml    = lane & 15;                // lane within half
    const int mt    = wave >> 2;                // m-tile (rows of KV)
    const int dt    = wave & 3;                 // d-tile (cols of KV)

    const int base = ((n * LSEQ) * NH + h) * DIM;   // + s*SSTRIDE + d

    v8f acc = {};
    float ksum_local = 0.0f;

    const int s_begin = split * SPAN;
    for (int c = 0; c < SPAN; c += SCHUNK) {
        // --- cooperative stage: 64 rows x 64 cols of K (with elu+1) and V ---
        for (int i = tid; i < SCHUNK * (DIM / 4); i += 512) {
            const int row = i >> 4;
            const int c4  = (i & 15) << 2;
            const int goff = base + (s_begin + c + row) * SSTRIDE + c4;
            float4 kq = *(const float4*)(Kg + goff);
            float4 vq = *(const float4*)(Vg + goff);
            kq.x = elu1(kq.x); kq.y = elu1(kq.y);
            kq.z = elu1(kq.z); kq.w = elu1(kq.w);
            *(float4*)&Kf[row][c4] = kq;
            *(float4*)&Vr[row][c4] = vq;
        }
        __syncthreads();

        // --- WMMA over the chunk: D[m,d] += A[m,k] * B[k,d], k = s-chunk ---
        #pragma unroll
        for (int kk = 0; kk < SCHUNK; kk += 4) {
            v2f a, b;
            a.x = Vr[kk + 2 * g + 0][16 * mt + ml];   // A[m][k] = V[s][m]
            a.y = Vr[kk + 2 * g + 1][16 * mt + ml];
            b.x = Kf[kk + 2 * g + 0][16 * dt + ml];   // B[k][d] = phi(K)[s][d]
            b.y = Kf[kk + 2 * g + 1][16 * dt + ml];
            acc = __builtin_amdgcn_wmma_f32_16x16x4_f32(
                false, a, false, b, (short)0, acc, false, false);
        }

        // --- Ksum partial: thread t < 64 owns column d = t ---
        if (tid < DIM) {
            float s = 0.0f;
            for (int r = 0; r < SCHUNK; ++r) s += Kf[r][tid];
            ksum_local += s;
        }
        __syncthreads();
    }

    // --- accumulate partials into workspace ---
    float* kvp = kv_ws + pair * (DIM * DIM);
    #pragma unroll
    for (int r = 0; r < 8; ++r) {
        const int mrow = 16 * mt + r + 8 * g;     // C/D layout: M = r + 8*(lane/16)
        atomicAdd(&kvp[mrow * DIM + 16 * dt + ml], acc[r]);
    }
    if (tid < DIM) atomicAdd(&ksum_ws[pair * DIM + tid], ksum_local);
}

// ---------------------------------------------------------------------------
// Kernel 2: Out[l,m] = (sum_d phi(Q)[l,d] * KV[m,d]) * Z[l],
//           Z[l] = 1 / (phi(Q)[l] . Ksum + eps)
// grid = PAIRS * (LSEQ/64) blocks, 512 threads (16 waves); each block emits a
// 64x64 output tile, each wave a 16x16 sub-tile via 16 WMMAs.
// ---------------------------------------------------------------------------
__global__ __launch_bounds__(512) void attn_out_kernel(
    const float* __restrict__ Qg, const float* __restrict__ kv_ws,
    const float* __restrict__ ksum_ws, float* __restrict__ Og)
{
    __shared__ float Qf[64][66];     // pad 66: A-frag reads stride rows
    __shared__ float KVt[64][65];    // transposed KV, pad 65: scatter writes
    __shared__ float KsumL[DIM];
    __shared__ float Zrow[64];

    const int pair  = blockIdx.x >> 6;
    const int ltile = blockIdx.x & 63;
    const int n     = pair >> 4;
    const int h     = pair & 15;
    const int l0    = ltile * 64;
    const int tid   = threadIdx.x;
    const int wave  = tid >> 5;
    const int lane  = tid & 31;
    const int g     = lane >> 4;
    const int ml    = lane & 15;
    const int lt    = wave >> 2;     // l sub-tile
    const int mt    = wave & 3;      // m sub-tile

    const int base = ((n * LSEQ) * NH + h) * DIM;

    // --- stage phi(Q) tile ---
    for (int i = tid; i < 64 * (DIM / 4); i += 512) {
        const int row = i >> 4;
        const int c4  = (i & 15) << 2;
        const int goff = base + (l0 + row) * SSTRIDE + c4;
        float4 q = *(const float4*)(Qg + goff);
        Qf[row][c4 + 0] = elu1(q.x);
        Qf[row][c4 + 1] = elu1(q.y);
        Qf[row][c4 + 2] = elu1(q.z);
        Qf[row][c4 + 3] = elu1(q.w);
    }
    // --- stage KV transposed: KVt[d][m] = KV[m][d] ---
    {
        const float* kvp = kv_ws + pair * (DIM * DIM);
        for (int i = tid; i < DIM * DIM; i += 512) {
            const int m = i >> 6;
            const int d = i & 63;
            KVt[d][m] = kvp[i];
        }
    }
    if (tid < DIM) KsumL[tid] = ksum_ws[pair * DIM + tid];
    __syncthreads();

    // --- per-row normalizer ---
    if (tid < 64) {
        float zacc = 1e-6f;
        #pragma unroll 8
        for (int d = 0; d < DIM; ++d) zacc += Qf[tid][d] * KsumL[d];
        Zrow[tid] = 1.0f / zacc;
    }
    __syncthreads();

    // --- WMMA: Out_tile = phi(Q)_tile (16x64) x KVt (64x16) ---
    v8f acc = {};
    #pragma unroll
    for (int kk = 0; kk < DIM; kk += 4) {
        v2f a, b;
        a.x = Qf[16 * lt + ml][kk + 2 * g + 0];   // A[l][k] = phi(Q)[l][d]
        a.y = Qf[16 * lt + ml][kk + 2 * g + 1];
        b.x = KVt[kk + 2 * g + 0][16 * mt + ml];  // B[k][m] = KV[m][d]
        b.y = KVt[kk + 2 * g + 1][16 * mt + ml];
        acc = __builtin_amdgcn_wmma_f32_16x16x4_f32(
            false, a, false, b, (short)0, acc, false, false);
    }

    // --- scale by Z and store (out[n][l][h][m], m contiguous) ---
    #pragma unroll
    for (int r = 0; r < 8; ++r) {
        const int lrow = 16 * lt + r + 8 * g;
        const float val = acc[r] * Zrow[lrow];
        Og[base + (l0 + lrow) * SSTRIDE + 16 * mt + ml] = val;
    }
}

extern "C" void kernel_launch(void* const* d_in, const int* in_sizes, int n_in,
                              void* d_out, int out_size, void* d_ws, size_t ws_size,
                              hipStream_t stream) {
    (void)in_sizes; (void)n_in; (void)out_size; (void)ws_size;
    const float* q = (const float*)d_in[0];
    const float* k = (const float*)d_in[1];
    const float* v = (const float*)d_in[2];
    float* out = (float*)d_out;

    float* kv_ws   = (float*)d_ws;                    // PAIRS * 64 * 64
    float* ksum_ws = kv_ws + PAIRS * DIM * DIM;       // PAIRS * 64
    const int ws_elems = PAIRS * DIM * DIM + PAIRS * DIM;

    zero_ws_kernel<<<(ws_elems + 255) / 256, 256, 0, stream>>>((float*)d_ws, ws_elems);
    kv_reduce_kernel<<<PAIRS * SPLIT, 512, 0, stream>>>(k, v, kv_ws, ksum_ws);
    attn_out_kernel<<<PAIRS * (LSEQ / 64), 512, 0, stream>>>(q, kv_ws, ksum_ws, out);
}